// Absolute_attention_89824946029339
// MI455X (gfx1250) — compile-verified
//
#include <hip/hip_runtime.h>
#include <math.h>

typedef __attribute__((ext_vector_type(2))) float v2f;
typedef __attribute__((ext_vector_type(8))) float v8f;

#define DIMN    256
#define NHEADS  16
#define HD      16
#define TDIM    64
#define TD2     128              // 2*TIME_DIM
#define BB      2
#define LL      2048
#define ROWS    (BB*LL)          // 4096
#define TSTRIDE (NHEADS*TD2)     // 2048 floats per position in the time table

// -------- fp32 WMMA helper: D = A(16x4) * B(4x16) + C --------
__device__ __forceinline__ v8f wmma_f32_k4(v2f a, v2f b, v8f c) {
  // 8 args: (neg_a, A, neg_b, B, c_mod, C, reuse_a, reuse_b)
  return __builtin_amdgcn_wmma_f32_16x16x4_f32(false, a, false, b, (short)0, c,
                                               false, false);
}

// -------- 1) time table: time[l, h, 0:64]=(c+s)/4 ; [64:128]=(c-s)/4 --------
__global__ void time_table_kernel(const float* __restrict__ time_angle,
                                  const float* __restrict__ head_delta,
                                  float* __restrict__ ttab) {
  int idx = blockIdx.x * blockDim.x + threadIdx.x;   // [0, L*H*64)
  int t = idx & (TDIM - 1);
  int h = (idx >> 6) & (NHEADS - 1);
  int l = idx >> 10;
  float ang = ((float)l + head_delta[h]) * time_angle[h * TDIM + t];
  float s, c;
  sincosf(ang, &s, &c);
  float* base = ttab + (size_t)l * TSTRIDE + h * TD2;
  base[t]        = (c + s) * 0.25f;   // inv_sqrt(HD) = 0.25 exactly
  base[t + TDIM] = (c - s) * 0.25f;
}

// -------- 2) generic row-major GEMM  C[M,256] = A[M,256] @ W[256,256] + bias --------
// one wave per 16x16 tile; K-loop of 4 per v_wmma_f32_16x16x4_f32
__global__ void wmma_gemm_rowmajor_256(const float* __restrict__ A,
                                       const float* __restrict__ W,
                                       const float* __restrict__ bias,
                                       float* __restrict__ C, int M) {
  const int K = DIMN, N = DIMN;
  int lane = threadIdx.x & 31;
  int wave = threadIdx.x >> 5;
  int tile = blockIdx.x * (blockDim.x >> 5) + wave;
  const int tilesN = N / 16;
  int tm = tile / tilesN;
  int tn = tile % tilesN;
  if (tm * 16 >= M) return;                 // wave-uniform; tile counts exact anyway
  int half = lane >> 4;                     // 0: lanes 0-15, 1: lanes 16-31
  int l16  = lane & 15;
  const float* Arow = A + (size_t)(tm * 16 + l16) * K;
  const float* Bcol = W + tn * 16 + l16;
  v8f acc = {};
  for (int k = 0; k < K; k += 4) {
    v2f a, b;
    int k0 = k + 2 * half;
    a.x = Arow[k0];                         // A: VGPR0=K{0,2}, VGPR1=K{1,3}
    a.y = Arow[k0 + 1];
    b.x = Bcol[(size_t)k0 * N];             // B: mirrored row striping
    b.y = Bcol[(size_t)(k0 + 1) * N];
    acc = wmma_f32_k4(a, b, acc);
  }
  int col = tn * 16 + l16;
  float bs = bias[col];
#pragma unroll
  for (int r = 0; r < 8; ++r) {             // D: VGPR r holds M=r / M=r+8
    int row = tm * 16 + r + 8 * half;
    C[(size_t)row * N + col] = acc[r] + bs;
  }
}

// -------- 3) qsum[row,h] = (1/16) * sum_d sigmoid(qlin[row, h*16+d]) --------
__global__ void qsum_kernel(const float* __restrict__ qlin,
                            float* __restrict__ qsum) {
  int idx = blockIdx.x * blockDim.x + threadIdx.x;   // [0, ROWS*NHEADS)
  int h   = idx & 15;
  int row = idx >> 4;
  const float* p = qlin + (size_t)row * DIMN + h * HD;
  float s = 0.f;
#pragma unroll
  for (int d = 0; d < HD; ++d) s += 1.0f / (1.0f + expf(-p[d]));
  qsum[idx] = s * (1.0f / HD);
}

// -------- 4) S[b,h] = T_h^T (128x2048) @ V[b,h] (2048x16) --------
__global__ void wmma_time_tv_kernel(const float* __restrict__ ttab,
                                    const float* __restrict__ v,
                                    float* __restrict__ S) {
  int lane = threadIdx.x & 31;
  int wave = threadIdx.x >> 5;
  int job  = blockIdx.x * (blockDim.x >> 5) + wave;  // [0, B*H*8)
  int tm = job & 7;
  int h  = (job >> 3) & 15;
  int b  = job >> 7;
  int half = lane >> 4, l16 = lane & 15;
  const float* Acol = ttab + h * TD2 + tm * 16 + l16;            // + kk*TSTRIDE
  const float* Bcol = v + (size_t)b * LL * DIMN + h * HD + l16;  // + kk*DIMN
  v8f acc = {};
  for (int kk = 0; kk < LL; kk += 4) {
    v2f a, bf;
    int k0 = kk + 2 * half;
    a.x  = Acol[(size_t)k0 * TSTRIDE];
    a.y  = Acol[(size_t)(k0 + 1) * TSTRIDE];
    bf.x = Bcol[(size_t)k0 * DIMN];
    bf.y = Bcol[(size_t)(k0 + 1) * DIMN];
    acc = wmma_f32_k4(a, bf, acc);
  }
  float* out = S + (size_t)((b * NHEADS + h) * TD2) * HD;
#pragma unroll
  for (int r = 0; r < 8; ++r) {
    int m = tm * 16 + r + 8 * half;
    out[(size_t)m * HD + l16] = acc[r];
  }
}

// -------- 5) acomb[b,l,h,:] = qsum[b,l,h] * (T_h (2048x128) @ S[b,h] (128x16)) --------
__global__ void wmma_ts_kernel(const float* __restrict__ ttab,
                               const float* __restrict__ S,
                               const float* __restrict__ qsum,
                               float* __restrict__ acomb) {
  int lane = threadIdx.x & 31;
  int wave = threadIdx.x >> 5;
  int job  = blockIdx.x * (blockDim.x >> 5) + wave;  // [0, B*H*(L/16)) = 4096
  int tm = job & 127;
  int h  = (job >> 7) & 15;
  int b  = job >> 11;
  int half = lane >> 4, l16 = lane & 15;
  const float* Arow  = ttab + (size_t)(tm * 16 + l16) * TSTRIDE + h * TD2;
  const float* Bbase = S + (size_t)((b * NHEADS + h) * TD2) * HD + l16;
  v8f acc = {};
  for (int kk = 0; kk < TD2; kk += 4) {
    v2f a, bf;
    int k0 = kk + 2 * half;
    a.x  = Arow[k0];
    a.y  = Arow[k0 + 1];
    bf.x = Bbase[(size_t)k0 * HD];
    bf.y = Bbase[(size_t)(k0 + 1) * HD];
    acc = wmma_f32_k4(a, bf, acc);
  }
#pragma unroll
  for (int r = 0; r < 8; ++r) {
    int l = tm * 16 + r + 8 * half;
    size_t grow = (size_t)b * LL + l;
    float sc = qsum[grow * NHEADS + h];
    acomb[grow * DIMN + h * HD + l16] = acc[r] * sc;
  }
}

// -------- 6) LayerNorm over last dim (256), one wave per row --------
__global__ void layernorm_kernel(const float* __restrict__ x,
                                 const float* __restrict__ gamma,
                                 const float* __restrict__ beta,
                                 float* __restrict__ out) {
  int lane = threadIdx.x & 31;
  int wave = threadIdx.x >> 5;
  int row  = blockIdx.x * (blockDim.x >> 5) + wave;
  const float* xr = x + (size_t)row * DIMN;
  float vals[8];
  float s = 0.f, sq = 0.f;
#pragma unroll
  for (int i = 0; i < 8; ++i) {
    float v = xr[lane + 32 * i];
    vals[i] = v;
    s += v;
    sq += v * v;
  }
#pragma unroll
  for (int off = 16; off > 0; off >>= 1) {
    s  += __shfl_xor(s, off, 32);
    sq += __shfl_xor(sq, off, 32);
  }
  float mean = s * (1.0f / DIMN);
  float var  = sq * (1.0f / DIMN) - mean * mean;   // population variance (ddof=0)
  float rstd = rsqrtf(var + 1e-5f);
  float* orow = out + (size_t)row * DIMN;
#pragma unroll
  for (int i = 0; i < 8; ++i) {
    int c = lane + 32 * i;
    orow[c] = (vals[i] - mean) * rstd * gamma[c] + beta[c];
  }
}

extern "C" void kernel_launch(void* const* d_in, const int* in_sizes, int n_in,
                              void* d_out, int out_size, void* d_ws, size_t ws_size,
                              hipStream_t stream) {
  const float* X      = (const float*)d_in[0];
  // d_in[1] attention_mask: only multiplies dead `k` -> unused
  const float* Wq     = (const float*)d_in[2];
  const float* bq     = (const float*)d_in[3];
  // d_in[4] Wk, d_in[5] bk: dead code in reference
  const float* Wv     = (const float*)d_in[6];
  const float* bv     = (const float*)d_in[7];
  const float* Wo     = (const float*)d_in[8];
  const float* bo     = (const float*)d_in[9];
  const float* tang   = (const float*)d_in[10];
  const float* hdel   = (const float*)d_in[11];
  const float* gamma  = (const float*)d_in[12];
  const float* beta   = (const float*)d_in[13];
  float* out = (float*)d_out;

  // workspace layout (floats), ~32.5 MB total
  float* w     = (float*)d_ws;
  float* ttab  = w;                                  // L*H*128      = 4,194,304
  float* qlin  = ttab  + (size_t)LL * TSTRIDE;       // ROWS*256     = 1,048,576
  float* vbuf  = qlin  + (size_t)ROWS * DIMN;        // ROWS*256     = 1,048,576
  float* qsum  = vbuf  + (size_t)ROWS * DIMN;        // ROWS*16      =    65,536
  float* Sbuf  = qsum  + (size_t)ROWS * NHEADS;      // B*H*128*16   =    65,536
  float* acomb = Sbuf  + (size_t)BB * NHEADS * TD2 * HD;  // ROWS*256
  float* outp  = acomb + (size_t)ROWS * DIMN;        // ROWS*256

  // 1) time table
  time_table_kernel<<<(LL * NHEADS * TDIM) / 256, 256, 0, stream>>>(tang, hdel, ttab);
  // 2) projections (4096x256x256 GEMMs, fp32 WMMA)
  wmma_gemm_rowmajor_256<<<512, 256, 0, stream>>>(X, Wq, bq, qlin, ROWS);
  wmma_gemm_rowmajor_256<<<512, 256, 0, stream>>>(X, Wv, bv, vbuf, ROWS);
  // 3) per-head sigmoid-mean
  qsum_kernel<<<(ROWS * NHEADS) / 256, 256, 0, stream>>>(qlin, qsum);
  // 4) S = T^T V  (per (b,h): 128x16, K=2048)
  wmma_time_tv_kernel<<<(BB * NHEADS * 8) / 8, 256, 0, stream>>>(ttab, vbuf, Sbuf);
  // 5) acomb = qsum * (T S)  (per (b,h): 2048x16, K=128)
  wmma_ts_kernel<<<(BB * NHEADS * (LL / 16)) / 8, 256, 0, stream>>>(ttab, Sbuf, qsum, acomb);
  // 6) output projection + LayerNorm
  wmma_gemm_rowmajor_256<<<512, 256, 0, stream>>>(acomb, Wo, bo, outp, ROWS);
  layernorm_kernel<<<ROWS / 8, 256, 0, stream>>>(outp, gamma, beta, out);
}